// SubLightGCNLayer_47425028882705
// MI455X (gfx1250) — compile-verified
//
#include <hip/hip_runtime.h>
#include <cstddef>

// SubLightGCN facility-location layer for MI455X (gfx1250, wave32).
// One wave32 workgroup per destination; 32x32 Gram via V_WMMA_F32_16X16X4_F32.

typedef float v2f __attribute__((ext_vector_type(2)));
typedef float v8f __attribute__((ext_vector_type(8)));

#define NBR 32
#define FD 64
#define KSEL 8
#define MS 68   // mail LDS row stride (floats): bank = (4*row + col) % 64 -> conflict-free b64 operand loads
#define SS 34   // sims LDS row stride (floats): half-wave store halves land in disjoint bank ranges

__global__ void deg_count_kernel(const int* __restrict__ nbr,
                                 unsigned* __restrict__ deg, int total) {
  int i = blockIdx.x * 256 + threadIdx.x;
  if (i < total) atomicAdd(&deg[nbr[i]], 1u);
}

__global__ __launch_bounds__(32) void facility_kernel(
    const float* __restrict__ h_src, const int* __restrict__ nbr,
    const unsigned* __restrict__ deg, float* __restrict__ out, int n_dst) {
  __shared__ float mail[NBR * MS];
  __shared__ float sims[NBR * SS];
  __shared__ float sqv[NBR];
  __shared__ float cachev[NBR];

  const int b = blockIdx.x;
  const int l = threadIdx.x;
  if (b >= n_dst) return;  // block-uniform; never taken with exact grid

  // ---- gather + out-degree^-1/2 scale + row squared-norms ----
  const int idx = nbr[(size_t)b * NBR + l];
  const unsigned dg = deg[idx];
  const float scale = 1.0f / sqrtf((float)(dg > 1u ? dg : 1u));
  const float* src = h_src + (size_t)idx * FD;
  float s = 0.0f;
#pragma unroll
  for (int c = 0; c < FD; c += 4) {
    float4 v = *(const float4*)(src + c);
    v.x *= scale; v.y *= scale; v.z *= scale; v.w *= scale;
    *(float4*)(&mail[l * MS + c]) = v;
    s += v.x * v.x + v.y * v.y + v.z * v.z + v.w * v.w;
  }
  sqv[l] = s;
  __syncthreads();

  const int n  = l & 15;
  const int hi = l >> 4;

  // ---- Gram matrix G = M * M^T (32x32) via 4 tiles of 16x16, K=64 ----
  // f32 16x16x4 operand layout: lane (l&15) -> row of its block; VGPR pair holds
  // K = {k+2*hi, k+2*hi+1}.  A-operand of row-block t == B-operand of col-block t.
  v8f c00 = {}; v8f c01 = {}; v8f c10 = {}; v8f c11 = {};
#pragma unroll
  for (int k = 0; k < FD; k += 4) {
    v2f x0 = *(const v2f*)(&mail[(n)      * MS + k + 2 * hi]);
    v2f x1 = *(const v2f*)(&mail[(16 + n) * MS + k + 2 * hi]);
    c00 = __builtin_amdgcn_wmma_f32_16x16x4_f32(false, x0, false, x0, (short)0, c00, false, false);
    c01 = __builtin_amdgcn_wmma_f32_16x16x4_f32(false, x0, false, x1, (short)0, c01, false, false);
    c10 = __builtin_amdgcn_wmma_f32_16x16x4_f32(false, x1, false, x0, (short)0, c10, false, false);
    c11 = __builtin_amdgcn_wmma_f32_16x16x4_f32(false, x1, false, x1, (short)0, c11, false, false);
  }

  // ---- dists = sqrt(clip(sq_i + sq_j - 2G)); mean; sims = exp(-d/denom) ----
  // C layout: tile(ti,tj), VGPR v, lane -> G[ti*16 + v + 8*hi][tj*16 + n]
  const float sqc0 = sqv[n];
  const float sqc1 = sqv[16 + n];
  float d00[8], d01[8], d10[8], d11[8];
  float lsum = 0.0f;
#pragma unroll
  for (int v = 0; v < 8; ++v) {
    const float sr0 = sqv[v + 8 * hi];
    const float sr1 = sqv[16 + v + 8 * hi];
    float t;
    t = sqrtf(fmaxf(sr0 + sqc0 - 2.0f * c00[v], 0.0f)); d00[v] = t; lsum += t;
    t = sqrtf(fmaxf(sr0 + sqc1 - 2.0f * c01[v], 0.0f)); d01[v] = t; lsum += t;
    t = sqrtf(fmaxf(sr1 + sqc0 - 2.0f * c10[v], 0.0f)); d10[v] = t; lsum += t;
    t = sqrtf(fmaxf(sr1 + sqc1 - 2.0f * c11[v], 0.0f)); d11[v] = t; lsum += t;
  }
#pragma unroll
  for (int off = 16; off > 0; off >>= 1) lsum += __shfl_xor(lsum, off, 32);
  const float mean  = lsum * (1.0f / (float)(NBR * NBR));
  const float denom = 1.0f /*SIGMA*/ * mean;
  const float r     = -1.0f / denom;
#pragma unroll
  for (int v = 0; v < 8; ++v) {
    sims[(v + 8 * hi)      * SS + n]       = expf(d00[v] * r);
    sims[(v + 8 * hi)      * SS + 16 + n]  = expf(d01[v] * r);
    sims[(16 + v + 8 * hi) * SS + n]       = expf(d10[v] * r);
    sims[(16 + v + 8 * hi) * SS + 16 + n]  = expf(d11[v] * r);
  }
  cachev[l] = 0.0f;
  float cache_l = 0.0f;
  __syncthreads();

  // ---- greedy facility-location selection (K rounds) ----
  int sels[KSEL];
#pragma unroll
  for (int k = 0; k < KSEL; ++k) {
    float gain = 0.0f;  // lane l owns candidate row l
#pragma unroll
    for (int j = 0; j < NBR; ++j) {
      const float cj = cachev[j];
      const float sv = sims[l * SS + j];
      gain += fmaxf(sv, cj) - cj;
    }
    // argmax across wave, first-index tie-break (matches jnp.argmax)
    float bg = gain; int bi = l;
#pragma unroll
    for (int off = 16; off > 0; off >>= 1) {
      const float og = __shfl_xor(bg, off, 32);
      const int   oi = __shfl_xor(bi, off, 32);
      if (og > bg || (og == bg && oi < bi)) { bg = og; bi = oi; }
    }
    sels[k] = bi;
    const float srow = sims[bi * SS + l];
    __syncthreads();            // all gain reads of cachev done
    cache_l = fmaxf(cache_l, srow);
    cachev[l] = cache_l;
    __syncthreads();            // cachev visible for next round
  }

  // ---- sum selected rows, in-degree normalization ----
  float acc0 = 0.0f, acc1 = 0.0f;
#pragma unroll
  for (int k = 0; k < KSEL; ++k) {
    acc0 += mail[sels[k] * MS + l];
    acc1 += mail[sels[k] * MS + 32 + l];
  }
  const float rn = 0.17677669529663687f;  // 32^-0.5
  out[(size_t)b * FD + l]      = acc0 * rn;
  out[(size_t)b * FD + 32 + l] = acc1 * rn;
}

extern "C" void kernel_launch(void* const* d_in, const int* in_sizes, int n_in,
                              void* d_out, int out_size, void* d_ws, size_t ws_size,
                              hipStream_t stream) {
  const float* h_src = (const float*)d_in[0];
  const int*   nbr   = (const int*)d_in[1];
  float*       out   = (float*)d_out;

  const int n_src = in_sizes[0] / FD;
  const int total = in_sizes[1];          // n_dst * NBR
  const int n_dst = total / NBR;

  unsigned* deg = (unsigned*)d_ws;        // n_src counters (~200 KB)
  hipMemsetAsync(deg, 0, (size_t)n_src * sizeof(unsigned), stream);
  deg_count_kernel<<<(total + 255) / 256, 256, 0, stream>>>(nbr, deg, total);
  facility_kernel<<<n_dst, 32, 0, stream>>>(h_src, nbr, deg, out, n_dst);
}